// BoundaryLoss_15934328668328
// MI455X (gfx1250) — compile-verified
//
#include <hip/hip_runtime.h>
#include <stdint.h>

// Problem constants from the reference
#define B_   4
#define C_   19
#define H_   256
#define W_   256
#define HW_  (H_ * W_)
#define NTOT (B_ * C_ * H_ * W_)          // 4,980,736
#define INF16 0x7FFF                       // vertical-distance sentinel (real max is 255)

typedef float v2f __attribute__((ext_vector_type(2)));
typedef float v8f __attribute__((ext_vector_type(8)));

// ---------------------------------------------------------------------------
// K1: vertical L1 distance scans (down + up) per (b, c, column).
// gpos[i][j] = vertical distance to nearest pixel of class c in column j
// gneg[i][j] = vertical distance to nearest pixel NOT of class c
// Exact integers; 0x7FFF means "no source in this column".
// ---------------------------------------------------------------------------
__global__ void vert_pass(const int* __restrict__ target,
                          unsigned short* __restrict__ gpos,
                          unsigned short* __restrict__ gneg) {
  int tid = blockIdx.x * blockDim.x + threadIdx.x;   // [0, B*C*W)
  if (tid >= B_ * C_ * W_) return;
  int j  = tid % W_;
  int bc = tid / W_;
  int c  = bc % C_;
  int b  = bc / C_;

  const int* tcol = target + (size_t)b * HW_ + j;
  unsigned short* pp = gpos + (size_t)bc * HW_ + j;
  unsigned short* pn = gneg + (size_t)bc * HW_ + j;

  int dp = INF16, dn = INF16;
  for (int i = 0; i < H_; ++i) {
    int p = (tcol[(size_t)i * W_] == c);
    dp = p ? 0 : dp + 1;
    dn = p ? dn + 1 : 0;
    pp[(size_t)i * W_] = (unsigned short)(dp > INF16 ? INF16 : dp);
    pn[(size_t)i * W_] = (unsigned short)(dn > INF16 ? INF16 : dn);
  }
  dp = INF16; dn = INF16;
  for (int i = H_ - 1; i >= 0; --i) {
    int p = (tcol[(size_t)i * W_] == c);
    dp = p ? 0 : dp + 1;
    dn = p ? dn + 1 : 0;
    int cp = dp > INF16 ? INF16 : dp;
    int cn = dn > INF16 ? INF16 : dn;
    unsigned short op = pp[(size_t)i * W_];
    unsigned short on = pn[(size_t)i * W_];
    pp[(size_t)i * W_] = (unsigned short)(cp < (int)op ? cp : (int)op);
    pn[(size_t)i * W_] = (unsigned short)(cn < (int)on ? cn : (int)on);
  }
}

// ---------------------------------------------------------------------------
// K2: per-pixel softmax statistics over the C=19 channels:
//     smax = max_c out[c], sinv = 1 / sum_c exp(out[c] - smax)
// ---------------------------------------------------------------------------
__global__ void softmax_stats(const float* __restrict__ out,
                              float* __restrict__ smax,
                              float* __restrict__ sinv) {
  int pix = blockIdx.x * blockDim.x + threadIdx.x;   // [0, B*H*W)
  if (pix >= B_ * HW_) return;
  int b = pix / HW_;
  int r = pix % HW_;
  const float* p = out + (size_t)b * C_ * HW_ + r;
  float m = p[0];
  for (int c = 1; c < C_; ++c) m = fmaxf(m, p[(size_t)c * HW_]);
  float s = 0.0f;
  for (int c = 0; c < C_; ++c) s += __expf(p[(size_t)c * HW_] - m);
  smax[pix] = m;
  sinv[pix] = 1.0f / s;
}

// ---------------------------------------------------------------------------
// K3: per (b, c, 64-row chunk): async-stage the uint16 vertical-distance tile
// into LDS, run the exact Felzenszwalb-Huttenlocher lower envelope per row,
// fuse dc with softmax probs, reduce the wave with v_wmma_f32_16x16x4_f32,
// atomicAdd into the scalar loss.
// ---------------------------------------------------------------------------
#define QROWS 64
__launch_bounds__(QROWS)
__global__ void fh_combine(const float* __restrict__ out,
                           const int* __restrict__ target,
                           const unsigned short* __restrict__ gpos,
                           const unsigned short* __restrict__ gneg,
                           const float* __restrict__ smax,
                           const float* __restrict__ sinv,
                           float* __restrict__ dout) {
  __shared__ unsigned short tile[QROWS * W_];        // 32 KB
  const int t    = threadIdx.x;                      // one row per thread
  const int quad = blockIdx.x & 3;
  const int bc   = blockIdx.x >> 2;
  const int c    = bc % C_;
  const int b    = bc / C_;
  const int gr   = quad * QROWS + t;                 // global row in image

  // per-thread scratch (private memory)
  short vst[W_];         // envelope parabola positions
  float zst[W_ + 1];     // envelope breakpoints
  float fvs[W_];         // f(v[k]) = squared vertical distance at v[k]

  float acc = 0.0f;

  for (int phase = 0; phase < 2; ++phase) {
    const unsigned short* gsrc = (phase == 0) ? gpos : gneg;

    // --- async DMA stage: 64 rows x 256 uint16 = 32 KB, 16B chunks ---------
    {
      const size_t gbase = ((size_t)bc * H_ + (size_t)quad * QROWS) * W_;
      uint64_t ga = (uint64_t)(uintptr_t)(const void*)(gsrc + gbase);
      uint32_t la = (uint32_t)(uintptr_t)(void*)(&tile[0]); // low 32b = LDS addr
      #pragma unroll
      for (int kk = 0; kk < 32; ++kk) {                 // 2048 chunks / 64 thr
        uint32_t off  = (uint32_t)((kk * QROWS + t) * 16);
        uint32_t ldst = la + off;
        uint64_t gsa  = ga + off;
        asm volatile("global_load_async_to_lds_b128 %0, %1, off"
                     :: "v"(ldst), "v"(gsa) : "memory");
      }
      asm volatile("s_wait_asynccnt 0" ::: "memory");
      __syncthreads();
    }

    // --- exact lower-envelope build on this thread's row --------------------
    const unsigned short* __restrict__ row = &tile[t * W_];
    int k = -1;
    for (int q = 0; q < W_; ++q) {
      unsigned short dv = row[q];
      if (dv >= (unsigned short)INF16) continue;        // column has no source
      float fq  = (float)((int)dv * (int)dv);
      float fqq = fq + (float)(q * q);
      if (k < 0) {
        k = 0; vst[0] = (short)q; zst[0] = -3.0e38f; zst[1] = 3.0e38f; fvs[0] = fq;
        continue;
      }
      float s;
      for (;;) {
        int p = vst[k];
        s = (fqq - (fvs[k] + (float)(p * p))) / (float)(2 * (q - p));
        if (s > zst[k]) break;
        --k;                                            // z[0] = -inf guards k
      }
      ++k;
      vst[k] = (short)q; zst[k] = s; zst[k + 1] = 3.0e38f; fvs[k] = fq;
    }

    // --- evaluate envelope, fuse with softmax prob, accumulate --------------
    if (k >= 0) {                                       // empty => class absent
      const size_t pixbase = (size_t)b * HW_ + (size_t)gr * W_;
      const float* orow = out + (size_t)bc * HW_ + (size_t)gr * W_;
      int ke = 0;
      for (int j = 0; j < W_; ++j) {
        while (zst[ke + 1] < (float)j) ++ke;
        int p = vst[ke];
        int tg = target[pixbase + j];
        bool hit = (phase == 0) ? (tg != c) : (tg == c);
        if (hit) {
          float d2   = (float)((j - p) * (j - p)) + fvs[ke];
          float dist = sqrtf(d2);
          float val  = (phase == 0) ? dist : (1.0f - dist);
          float pr   = __expf(orow[j] - smax[pixbase + j]) * sinv[pixbase + j];
          acc += pr * val;
        }
      }
    }
    __syncthreads();                                    // tile reused next phase
  }

  // --- wave reduction via WMMA: D = A(16x4) * ones(4x16); each lane's acc
  // sits in one A slot, so sum_m D[m, n] over the two half-columns (lane 0
  // and lane 16) equals the full 32-lane wave sum, in exact f32 FMA adds. ---
  v2f a;    a.x = acc * (1.0f / (float)NTOT); a.y = 0.0f;
  v2f ones; ones.x = 1.0f; ones.y = 1.0f;
  v8f cz = {};
  v8f d = __builtin_amdgcn_wmma_f32_16x16x4_f32(false, a, false, ones,
                                                (short)0, cz, false, false);
  float wsum = d[0] + d[1] + d[2] + d[3] + d[4] + d[5] + d[6] + d[7];
  int lane = t & 31;
  if (lane == 0 || lane == 16) atomicAdd(dout, wsum);
}

// ---------------------------------------------------------------------------
extern "C" void kernel_launch(void* const* d_in, const int* in_sizes, int n_in,
                              void* d_out, int out_size, void* d_ws, size_t ws_size,
                              hipStream_t stream) {
  const float* output = (const float*)d_in[0];
  const int*   target = (const int*)d_in[1];

  // Workspace layout (~22 MB total, 256B aligned slices)
  char* ws = (char*)d_ws;
  size_t off = 0;
  unsigned short* gpos = (unsigned short*)(ws + off);
  off += (size_t)NTOT * sizeof(unsigned short); off = (off + 255) & ~(size_t)255;
  unsigned short* gneg = (unsigned short*)(ws + off);
  off += (size_t)NTOT * sizeof(unsigned short); off = (off + 255) & ~(size_t)255;
  float* smax = (float*)(ws + off);
  off += (size_t)(B_ * HW_) * sizeof(float);    off = (off + 255) & ~(size_t)255;
  float* sinv = (float*)(ws + off);

  hipMemsetAsync(d_out, 0, sizeof(float), stream);

  {
    int n = B_ * C_ * W_;                      // 19456 columns
    vert_pass<<<(n + 255) / 256, 256, 0, stream>>>(target, gpos, gneg);
  }
  {
    int n = B_ * HW_;                          // 262144 pixels
    softmax_stats<<<(n + 255) / 256, 256, 0, stream>>>(output, smax, sinv);
  }
  {
    int nblk = B_ * C_ * 4;                    // 304 blocks x 64 threads
    fh_combine<<<nblk, QROWS, 0, stream>>>(output, target, gpos, gneg,
                                           smax, sinv, (float*)d_out);
  }
}